// HypGraphSelfAttentionLayer_54554674593976
// MI455X (gfx1250) — compile-verified
//
#include <hip/hip_runtime.h>
#include <hip/hip_bf16.h>
#include <math.h>

// ---------------- problem constants (from reference) ----------------
#define Bsz   8
#define Mm    32
#define Nn    64
#define Dd    2048            // M*N
#define OUTIN 65536           // M*M*N
#define MAXNORM 0.996f        // (1 - PROJ_EPS)/sqrt(c), c=1
#define MINN    1e-15f

// ---------------- workspace layout (floats). Needs ~2.5 MB ----------
#define OFF_XP   0                         // padded X: 16 x 2048 (rows 8..15 zero)
#define OFF_AV   (OFF_XP  + 16*Dd)         // padded att_vectors: 16 x 2048
#define OFF_MX   (OFF_AV  + 16*Dd)         // mx = X @ W_in^T : 8 x 65536
#define OFF_MX2  (OFF_MX  + Bsz*OUTIN)     // mx2 = AV @ W_out^T : 8 x 2048
#define OFF_KK   (OFF_MX2 + Bsz*Dd)        // klein coords: 8 x 32 x 64
#define OFF_GAM  (OFF_KK  + Bsz*Dd)        // gamma: 8 x 32
#define OFF_SCAL (OFF_GAM + Bsz*Mm)        // scalars (40 floats)

// scalar slots within OFF_SCAL
#define SC_XNORM 0    // 8: ||poincare_in row||
#define SC_MXSQ  8    // 8: sum mx^2 per row (atomic)
#define SC_AVSQ  16   // 8: sum att_vec^2 per row (atomic)
#define SC_MX2SQ 24   // 8: sum mx2^2 per row (atomic)
#define SC_F1    32   // 8: mobius scalar factor for GEMM1 output

typedef float v2f __attribute__((ext_vector_type(2)));
typedef float v8f __attribute__((ext_vector_type(8)));
typedef unsigned int v4u __attribute__((ext_vector_type(4)));
typedef int v4i __attribute__((ext_vector_type(4)));
typedef int v8i __attribute__((ext_vector_type(8)));

// TDM K-chunking for the big GEMM
#define CKDW  64                 // K-chunk per TDM transfer, in floats
#define ROWDW (CKDW + 4)         // LDS row stride after D# padding (4 DW / 64 DW)
#define BUFDW (16 * ROWDW)       // one LDS buffer: 16 rows, padded

// ---------------- helpers ----------------
__device__ inline float block_reduce_sum(float v, float* sbuf) {
  const int t = threadIdx.x;
  sbuf[t] = v;
  __syncthreads();
  for (int s = blockDim.x >> 1; s > 0; s >>= 1) {
    if (t < s) sbuf[t] += sbuf[t + s];
    __syncthreads();
  }
  float r = sbuf[0];
  __syncthreads();
  return r;
}

__device__ inline float wave_reduce_sum(float v) {
#pragma unroll
  for (int m = 16; m >= 1; m >>= 1) v += __shfl_xor(v, m, 32);
  return v;
}

__device__ inline float artanh_clip(float x) {
  x = fminf(fmaxf(x, -1.0f + 1e-7f), 1.0f - 1e-7f);
  return 0.5f * logf((1.0f + x) / (1.0f - x));
}

// Issue one TDM transfer: 2D tile (tile_dim0 = CKDW elems, tile_dim1 = 16 rows)
// of a row-major (Ncols x K) fp32 tensor into LDS at lds_off, with LDS padding
// of 4 DWORDs after every 64 DWORDs (row stride ROWDW).
__device__ inline void tdm_issue_tile(const float* gtile, unsigned lds_off,
                                      int K, int Ncols) {
  unsigned long long ga = (unsigned long long)(size_t)gtile;
  v4u g0;
  g0[0] = 1u;                                                // count=1, user mode
  g0[1] = lds_off;                                           // lds_addr (bytes)
  g0[2] = (unsigned)ga;                                      // global_addr[31:0]
  g0[3] = ((unsigned)(ga >> 32) & 0x01FFFFFFu) | (2u << 30); // addr[56:32] | type=2
  v8i g1;
  // data_size=2 (4B) | pad_enable | pad_interval=5 (64 DW) | pad_amount=3 (4 DW)
  g1[0] = (int)((2u << 16) | (1u << 20) | (5u << 22) | (3u << 25));
  g1[1] = (int)(((unsigned)K & 0xFFFFu) << 16);              // tensor_dim0[15:0]
  g1[2] = (int)(((unsigned)K >> 16) | (((unsigned)Ncols & 0xFFFFu) << 16));
  g1[3] = (int)(((unsigned)Ncols >> 16) | ((unsigned)CKDW << 16)); // tile_dim0
  g1[4] = 16;                                                // tile_dim1=16
  g1[5] = K;                                                 // tensor_dim0_stride[31:0]
  g1[6] = 0;
  g1[7] = 0;
  v4i z4 = {0, 0, 0, 0};
#if __clang_major__ >= 23
  v8i z8 = {0, 0, 0, 0, 0, 0, 0, 0};
  __builtin_amdgcn_tensor_load_to_lds(g0, g1, z4, z4, z8, 0);
#else
  __builtin_amdgcn_tensor_load_to_lds(g0, g1, z4, z4, 0);
#endif
}

// ---------------- stage 1: project inputs onto Poincare ball --------
__global__ void __launch_bounds__(256)
prep_proj(const float* __restrict__ x, float* __restrict__ Xp,
          float* __restrict__ AV, float* __restrict__ scal) {
  __shared__ float sbuf[256];
  const int b = blockIdx.x, t = threadIdx.x;
  if (b < Bsz) {
    float ss = 0.0f;
    for (int i = t; i < Dd; i += 256) { float v = x[b * Dd + i]; ss += v * v; }
    ss = block_reduce_sum(ss, sbuf);
    const float norm  = sqrtf(ss);
    const float scale = (norm > MAXNORM) ? (MAXNORM / norm) : 1.0f;
    for (int i = t; i < Dd; i += 256) Xp[b * Dd + i] = x[b * Dd + i] * scale;
    if (t == 0) scal[SC_XNORM + b] = fmaxf(fminf(norm, MAXNORM), MINN);
  } else {
    for (int i = t; i < Dd; i += 256) {
      Xp[b * Dd + i] = 0.0f;
      AV[b * Dd + i] = 0.0f;
    }
    if (b == 8 && t < 24) scal[SC_MXSQ + t] = 0.0f;  // zero the atomic accumulators
  }
}

// ---------------- TDM + WMMA GEMM (big streaming GEMM) --------------
// Y(8 x Ncols) = A16(16 x K) @ W(Ncols x K)^T. One wave = one 16x16 tile.
// W is staged into LDS by the Tensor Data Mover, double-buffered per wave,
// synchronized with s_wait_tensorcnt. Rows 8..15 of A16 are zero pad.
__global__ void __launch_bounds__(128)
gemm16_wmma_tdm(const float* __restrict__ A16, const float* __restrict__ W,
                float* __restrict__ Y, float* __restrict__ rowsq,
                int K, int Ncols) {
  __shared__ float smem[4 * 2 * BUFDW];
  const int lane = threadIdx.x & 31;
  const int wv   = threadIdx.x >> 5;
  const int wave = blockIdx.x * 4 + wv;
  const int col0 = wave << 4;
  if (col0 >= Ncols) return;              // wave-uniform
  const int half = lane >> 4;
  const int l16  = lane & 15;
  const float* __restrict__ xrow   = A16 + (size_t)l16 * K + 2 * half;
  const float* __restrict__ wtile0 = W + (size_t)col0 * K;
  float* lb0 = &smem[wv * 2 * BUFDW];
  float* lb1 = lb0 + BUFDW;
  const unsigned lds0 = (unsigned)(size_t)(void*)lb0;  // LDS byte offset
  const unsigned lds1 = (unsigned)(size_t)(void*)lb1;

  v8f acc = {};
  const int nchunk = K / CKDW;
  tdm_issue_tile(wtile0, lds0, K, Ncols);
  for (int c = 0; c < nchunk; ++c) {
    const int sel = c & 1;
    asm volatile("" ::: "memory");
    if (c + 1 < nchunk) {
      tdm_issue_tile(wtile0 + (c + 1) * CKDW, sel ? lds0 : lds1, K, Ncols);
      __builtin_amdgcn_s_wait_tensorcnt(1);   // chunk c landed in LDS
    } else {
      __builtin_amdgcn_s_wait_tensorcnt(0);
    }
    asm volatile("" ::: "memory");
    const float* bb = (sel ? lb1 : lb0) + l16 * ROWDW + 2 * half;
    const float* xb = xrow + c * CKDW;
#pragma unroll
    for (int kk = 0; kk < CKDW; kk += 4) {
      v2f a, b;
      a.x = xb[kk]; a.y = xb[kk + 1];
      b.x = bb[kk]; b.y = bb[kk + 1];      // conflict-free ds_load_b64 (padded rows)
      acc = __builtin_amdgcn_wmma_f32_16x16x4_f32(
          false, a, false, b, (short)0, acc, false, false);
    }
  }
  if (half == 0) {
#pragma unroll
    for (int r = 0; r < 8; ++r)
      Y[(size_t)r * Ncols + col0 + l16] = acc[r];
  }
#pragma unroll
  for (int r = 0; r < 8; ++r) {
    float s = wave_reduce_sum(acc[r] * acc[r]);
    if (lane == 0) atomicAdd(&rowsq[r], s);
  }
}

// ---------------- direct-load WMMA GEMM (small GEMM, L2 resident) ---
__global__ void __launch_bounds__(128)
gemm16_wmma(const float* __restrict__ A16, const float* __restrict__ W,
            float* __restrict__ Y, float* __restrict__ rowsq,
            int K, int Ncols) {
  const int lane = threadIdx.x & 31;
  const int wave = blockIdx.x * (blockDim.x >> 5) + (threadIdx.x >> 5);
  const int col0 = wave << 4;
  if (col0 >= Ncols) return;
  const int half = lane >> 4;
  const int l16  = lane & 15;
  const float* __restrict__ xrow = A16 + (size_t)l16 * K + 2 * half;
  const float* __restrict__ wrow = W + (size_t)(col0 + l16) * K + 2 * half;
  v8f acc = {};
  for (int k = 0; k < K; k += 64) {
    if (k + 512 < K) {
      __builtin_prefetch(wrow + k + 512, 0, 0);
    }
#pragma unroll
    for (int kk = 0; kk < 64; kk += 4) {
      v2f a, b;
      a.x = xrow[k + kk]; a.y = xrow[k + kk + 1];
      b.x = wrow[k + kk]; b.y = wrow[k + kk + 1];
      acc = __builtin_amdgcn_wmma_f32_16x16x4_f32(
          false, a, false, b, (short)0, acc, false, false);
    }
  }
  if (half == 0) {
#pragma unroll
    for (int r = 0; r < 8; ++r)
      Y[(size_t)r * Ncols + col0 + l16] = acc[r];
  }
#pragma unroll
  for (int r = 0; r < 8; ++r) {
    float s = wave_reduce_sum(acc[r] * acc[r]);
    if (lane == 0) atomicAdd(&rowsq[r], s);
  }
}

// ---------------- stage 3: per-row mobius_matvec scalar factor ------
__global__ void factor1_kernel(float* __restrict__ scal) {
  const int b = threadIdx.x;
  if (b < Bsz) {
    const float xn  = scal[SC_XNORM + b];
    const float mxn = fmaxf(sqrtf(scal[SC_MXSQ + b]), MINN);
    const float th  = tanhf(mxn / xn * artanh_clip(xn));
    float f = th / mxn;
    const float nr = fabsf(th);
    if (nr > MAXNORM) f *= MAXNORM / nr;      // fold proj()
    scal[SC_F1 + b] = f;
  }
}

// ---------------- stage 4: poincare -> klein per node, gamma --------
__global__ void __launch_bounds__(64)
klein_prep(const float* __restrict__ Xp, float* __restrict__ Kk,
           float* __restrict__ gam) {
  __shared__ float sbuf[64];
  const int b = blockIdx.x >> 5, j = blockIdx.x & 31, n = threadIdx.x;
  const float v   = Xp[(size_t)b * Dd + j * Nn + n];
  const float vsq = block_reduce_sum(v * v, sbuf);
  const float kk  = 2.0f * v / (1.0f + vsq);
  const float ksq = block_reduce_sum(kk * kk, sbuf);
  Kk[(size_t)b * Dd + j * Nn + n] = kk;
  if (n == 0) gam[b * Mm + j] = rsqrtf(fmaxf(1.0f - ksq, 1e-7f));
}

// ---------------- stage 5: masked Einstein midpoint per (b,i) -------
__global__ void __launch_bounds__(64)
midpoint_kernel(const float* __restrict__ MX, const float* __restrict__ Kk,
                const float* __restrict__ gam, const int* __restrict__ mask,
                float* __restrict__ scal, float* __restrict__ AV) {
  __shared__ float sbuf[64];
  const int b = blockIdx.x >> 5, i = blockIdx.x & 31, n = threadIdx.x;
  const float f = scal[SC_F1 + b];
  const float* __restrict__ mxb = MX + (size_t)b * OUTIN + (size_t)i * (Mm * Nn);
  const float* __restrict__ kb  = Kk + (size_t)b * Dd;
  float num = 0.0f, den = 0.0f;
  for (int j = 0; j < Mm; ++j) {
    if (mask[i * Mm + j] != 0) {
      const float wj = f * mxb[j * Nn + n] * gam[b * Mm + j];
      num += wj * kb[j * Nn + n];
      den += wj;
    }
  }
  if (fabsf(den) < MINN) den = MINN;
  const float r   = num / den;
  const float sqr = block_reduce_sum(r * r, sbuf);
  const float sqc = fminf(fmaxf(sqr, 0.0f), 1.0f - 1e-7f);
  const float inv = 1.0f / (1.0f + sqrtf(1.0f - sqc));
  float p   = r * inv;
  const float pn2 = sqr * inv * inv;
  const float pn  = sqrtf(pn2);
  const float s   = (pn > MAXNORM) ? (MAXNORM / pn) : 1.0f;
  p *= s;
  AV[(size_t)b * Dd + i * Nn + n] = p;
  if (n == 0) atomicAdd(&scal[SC_AVSQ + b], pn2 * s * s);
}

// ---------------- stage 7: mobius scalar for GEMM2 + hyperboloid ----
__global__ void __launch_bounds__(256)
final_kernel(const float* __restrict__ MX2, const float* __restrict__ scal,
             float* __restrict__ out) {
  const int b = blockIdx.x, t = threadIdx.x;
  const float xn  = fmaxf(sqrtf(scal[SC_AVSQ + b]), MINN);
  const float mxn = fmaxf(sqrtf(scal[SC_MX2SQ + b]), MINN);
  const float th  = tanhf(mxn / fminf(xn, MAXNORM) * artanh_clip(xn));
  float g = th / mxn;
  const float nr = fabsf(th);
  if (nr > MAXNORM) g *= MAXNORM / nr;
  const float sq = g * g * mxn * mxn;
  const float invden = 1.0f / (1.0f - sq);
  if (t == 0) out[b * (Dd + 1)] = (1.0f + sq) * invden;
  for (int d = t; d < Dd; d += 256)
    out[b * (Dd + 1) + 1 + d] = 2.0f * g * MX2[(size_t)b * Dd + d] * invden;
}

// ---------------- launcher ----------------
extern "C" void kernel_launch(void* const* d_in, const int* in_sizes, int n_in,
                              void* d_out, int out_size, void* d_ws, size_t ws_size,
                              hipStream_t stream) {
  const float* x     = (const float*)d_in[0];  // (8, 2048)
  const int*   mask  = (const int*)d_in[1];    // (32, 32)
  const float* W_in  = (const float*)d_in[2];  // (65536, 2048)
  const float* W_out = (const float*)d_in[3];  // (2048, 2048)
  float* out = (float*)d_out;                  // (8, 2049)
  float* ws  = (float*)d_ws;

  float* Xp   = ws + OFF_XP;
  float* AV   = ws + OFF_AV;
  float* MX   = ws + OFF_MX;
  float* MX2  = ws + OFF_MX2;
  float* Kk   = ws + OFF_KK;
  float* gam  = ws + OFF_GAM;
  float* scal = ws + OFF_SCAL;

  prep_proj<<<16, 256, 0, stream>>>(x, Xp, AV, scal);
  // GEMM1 (streams 512 MB of W_in via TDM): 4096 waves -> 1024 blocks x 4 waves
  gemm16_wmma_tdm<<<OUTIN / 16 / 4, 128, 0, stream>>>(Xp, W_in, MX,
                                                      scal + SC_MXSQ, Dd, OUTIN);
  factor1_kernel<<<1, 32, 0, stream>>>(scal);
  klein_prep<<<Bsz * Mm, 64, 0, stream>>>(Xp, Kk, gam);
  midpoint_kernel<<<Bsz * Mm, 64, 0, stream>>>(MX, Kk, gam, mask, scal, AV);
  // GEMM2 (16 MB, L2-resident): direct-load WMMA path
  gemm16_wmma<<<Dd / 16 / 4, 128, 0, stream>>>(AV, W_out, MX2, scal + SC_MX2SQ,
                                               Dd, Dd);
  final_kernel<<<Bsz, 256, 0, stream>>>(MX2, scal, out);
}